// POTLoss_35364760715834
// MI455X (gfx1250) — compile-verified
//
#include <hip/hip_runtime.h>
#include <stdint.h>

// Wasserstein-1D (p=2) loss: per-row sort of x and y, mean squared diff, sum.
// Strategy: per-row 4x8-bit LSD radix sort (one workgroup per (row, array)),
// with CDNA5 Tensor Data Mover staging input tiles into LDS (double-buffered,
// s_wait_tensorcnt) and wave32 ballot multisplit for stable in-pass ranks.
// Workspace layout (needs 512 MB):
//   bufA = ws + 0        : u32 keys [2][128][262144]
//   bufB = ws + 256 MB   : u32 keys / final sorted floats [2][128][262144]
//   per-row partial sums reuse the bufA region after sorting completes.

#define BATCH 128
#define NELEM 262144
#define TPB 256
#define TILE 4096
#define NTILES (NELEM / TILE)  // 64
#define CHUNKS (TILE / TPB)    // 16

typedef uint32_t v4u __attribute__((ext_vector_type(4)));
typedef int      v8i __attribute__((ext_vector_type(8)));
typedef int      v4i __attribute__((ext_vector_type(4)));

// float bits -> order-preserving unsigned key, and inverse
__device__ __forceinline__ uint32_t f2k(uint32_t u) {
  return u ^ (uint32_t)(((int32_t)u >> 31) | 0x80000000);
}
__device__ __forceinline__ uint32_t k2f(uint32_t k) {
  return k ^ (uint32_t)(((int32_t)(~k) >> 31) | 0x80000000);
}

// Issue one TDM 1-D tile copy: global (elems * 4 bytes) -> LDS at lds_off.
// D# per CDNA5 ISA §8.3-8.6: group0 {count=1, lds_addr, global_addr, type=2},
// group1 {data_size=4B, tensor_dim0=elems, tensor_dim1=1, tile_dim0=elems,
// tile_dim1=0 (1-D), dim0_stride=elems}, groups 2/3 zero (<=2-D tensor).
__device__ __forceinline__ void tdm_tile_to_lds(uint32_t lds_off, uint64_t gaddr,
                                                uint32_t elems) {
  v4u g0 = { 1u,                                   // count=1, user mode
             lds_off,                              // lds_addr (bytes)
             (uint32_t)gaddr,                      // global_addr[31:0]
             ((uint32_t)(gaddr >> 32) & 0x01FFFFFFu) | (2u << 30) }; // [56:32]|type=2
  v8i g1 = { (int)(2u << 16),                      // workgroup_mask=0, data_size=2 (4B)
             (int)((elems & 0xFFFFu) << 16),       // tensor_dim0[15:0]
             (int)(((elems >> 16) & 0xFFFFu) | (1u << 16)), // dim0[31:16] | tensor_dim1=1
             (int)((elems & 0xFFFFu) << 16),       // tile_dim0 (<=65535)
             0,                                    // tile_dim1=0, tile_dim2=0
             (int)elems,                           // tensor_dim0_stride[31:0]
             0, 0 };
  v4i g2 = { 0, 0, 0, 0 };
  v4i g3 = { 0, 0, 0, 0 };
#if defined(__clang_major__) && __clang_major__ >= 23
  v8i gx = { 0, 0, 0, 0, 0, 0, 0, 0 };             // extra group (zero-filled)
  __builtin_amdgcn_tensor_load_to_lds(g0, g1, g2, g3, gx, 0);
#else
  __builtin_amdgcn_tensor_load_to_lds(g0, g1, g2, g3, 0);
#endif
}

__global__ __launch_bounds__(TPB) void potws_sort_kernel(
    const float* __restrict__ x, const float* __restrict__ y,
    uint32_t* __restrict__ bufA, uint32_t* __restrict__ bufB) {
  __shared__ uint32_t kbuf[2][TILE];    // 32 KB TDM staging, double buffered
  __shared__ uint32_t hist4[4][256];    // all four digit histograms -> bases
  __shared__ uint32_t run[256];         // running bucket offsets (current pass)
  __shared__ uint32_t whist[8 * 256];   // per-wave digit counts per 256-chunk

  const int arr = blockIdx.x & 1;
  const int row = blockIdx.x >> 1;
  const float* srcf = (arr ? y : x) + (size_t)row * NELEM;
  uint32_t* A  = bufA + ((size_t)arr * BATCH + row) * (size_t)NELEM;
  uint32_t* Bb = bufB + ((size_t)arr * BATCH + row) * (size_t)NELEM;

  const int tid = threadIdx.x;
  const int lane = tid & 31;
  const int wav = tid >> 5;
  const uint32_t ltmask = (1u << lane) - 1u;
  const bool issuer = (tid < 32);       // wave 0 owns TDM issue + TENSORcnt

  const uint32_t lds0 = (uint32_t)(uintptr_t)(&kbuf[0][0]);
  const uint32_t lds1 = (uint32_t)(uintptr_t)(&kbuf[1][0]);

  // ---- Phase H: histograms of all 4 radix digits in one streamed read ----
  for (int i = tid; i < 4 * 256; i += TPB) ((uint32_t*)hist4)[i] = 0u;
  __syncthreads();

  const uint64_t sb = (uint64_t)(uintptr_t)srcf;
  if (issuer) tdm_tile_to_lds(lds0, sb, TILE);
  for (int t = 0; t < NTILES; ++t) {
    if (issuer) {
      if (t + 1 < NTILES) {
        tdm_tile_to_lds(((t + 1) & 1) ? lds1 : lds0,
                        sb + (uint64_t)(t + 1) * TILE * 4u, TILE);
        __builtin_amdgcn_s_wait_tensorcnt(1);
      } else {
        __builtin_amdgcn_s_wait_tensorcnt(0);
      }
    }
    __syncthreads();
    const uint32_t* kb = kbuf[t & 1];
    for (int i = 0; i < CHUNKS; ++i) {
      uint32_t k = f2k(kb[i * TPB + tid]);
      atomicAdd(&hist4[0][k & 255u], 1u);
      atomicAdd(&hist4[1][(k >> 8) & 255u], 1u);
      atomicAdd(&hist4[2][(k >> 16) & 255u], 1u);
      atomicAdd(&hist4[3][k >> 24], 1u);
    }
    __syncthreads();
  }

  // exclusive scan of each digit histogram -> bucket bases
  if (tid == 0) {
    for (int p = 0; p < 4; ++p) {
      uint32_t s = 0;
      for (int d = 0; d < 256; ++d) { uint32_t c = hist4[p][d]; hist4[p][d] = s; s += c; }
    }
  }
  __syncthreads();

  // ---- 4 stable LSD scatter passes: in->A->B->A->B ----
  for (int pass = 0; pass < 4; ++pass) {
    const bool first = (pass == 0), last = (pass == 3);
    const uint32_t* srck = first ? (const uint32_t*)srcf : ((pass == 2) ? Bb : A);
    uint32_t* dst = (pass & 1) ? Bb : A;
    const int sh = 8 * pass;

    run[tid] = hist4[pass][tid];
    __syncthreads();

    const uint64_t sp = (uint64_t)(uintptr_t)srck;
    if (issuer) tdm_tile_to_lds(lds0, sp, TILE);
    for (int t = 0; t < NTILES; ++t) {
      if (issuer) {
        if (t + 1 < NTILES) {
          tdm_tile_to_lds(((t + 1) & 1) ? lds1 : lds0,
                          sp + (uint64_t)(t + 1) * TILE * 4u, TILE);
          __builtin_amdgcn_s_wait_tensorcnt(1);
        } else {
          __builtin_amdgcn_s_wait_tensorcnt(0);
        }
      }
      __syncthreads();
      const uint32_t* kb = kbuf[t & 1];
      for (int c = 0; c < CHUNKS; ++c) {
        // zero per-wave digit counts
        for (int w = 0; w < 8; ++w) whist[w * 256 + tid] = 0u;
        __syncthreads();

        uint32_t k = kb[c * TPB + tid];
        if (first) k = f2k(k);
        const uint32_t d = (k >> sh) & 255u;

        // wave32 ballot multisplit: mask of lanes sharing this digit
        uint32_t mask = 0xFFFFFFFFu;
#pragma unroll
        for (int b = 0; b < 8; ++b) {
          uint32_t bal = __builtin_amdgcn_ballot_w32((d >> b) & 1u);
          mask &= ((d >> b) & 1u) ? bal : ~bal;
        }
        const uint32_t below = mask & ltmask;
        const uint32_t rank = (uint32_t)__popc(below);
        if (below == 0u) whist[wav * 256 + d] = (uint32_t)__popc(mask);
        __syncthreads();

        uint32_t before = 0;
        for (int w = 0; w < wav; ++w) before += whist[w * 256 + d];
        const uint32_t pos = run[d] + before + rank;
        dst[pos] = last ? k2f(k) : k;   // final pass stores sorted floats
        __syncthreads();

        uint32_t tot = 0;
#pragma unroll
        for (int w = 0; w < 8; ++w) tot += whist[w * 256 + tid];
        run[tid] += tot;
        __syncthreads();
      }
    }
    __threadfence();   // scattered stores visible before next pass's TDM reads
    __syncthreads();
  }
}

__global__ __launch_bounds__(TPB) void potws_cost_kernel(
    const float* __restrict__ sorted, float* __restrict__ partial) {
  __shared__ double red[TPB];
  const int row = blockIdx.x;
  const float4* xs = (const float4*)(sorted + (size_t)row * NELEM);
  const float4* ys = (const float4*)(sorted + ((size_t)BATCH + row) * NELEM);
  double acc = 0.0;
  for (int i = threadIdx.x; i < NELEM / 4; i += TPB) {
    float4 a = xs[i]; float4 b = ys[i];
    float d0 = a.x - b.x, d1 = a.y - b.y, d2 = a.z - b.z, d3 = a.w - b.w;
    acc += (double)d0 * d0 + (double)d1 * d1 + (double)d2 * d2 + (double)d3 * d3;
  }
  red[threadIdx.x] = acc;
  __syncthreads();
  for (int s = TPB / 2; s > 0; s >>= 1) {
    if (threadIdx.x < (unsigned)s) red[threadIdx.x] += red[threadIdx.x + s];
    __syncthreads();
  }
  if (threadIdx.x == 0) partial[row] = (float)(red[0] / (double)NELEM);
}

__global__ void potws_final_kernel(const float* __restrict__ partial,
                                   float* __restrict__ out) {
  __shared__ double red[BATCH];
  const int t = threadIdx.x;
  red[t] = (double)partial[t];
  __syncthreads();
  for (int s = BATCH / 2; s > 0; s >>= 1) {
    if (t < s) red[t] += red[t + s];
    __syncthreads();
  }
  if (t == 0) out[0] = (float)red[0];
}

extern "C" void kernel_launch(void* const* d_in, const int* in_sizes, int n_in,
                              void* d_out, int out_size, void* d_ws, size_t ws_size,
                              hipStream_t stream) {
  (void)in_sizes; (void)n_in; (void)out_size; (void)ws_size;
  const float* x = (const float*)d_in[0];
  const float* y = (const float*)d_in[1];
  uint32_t* bufA = (uint32_t*)d_ws;
  uint32_t* bufB = bufA + (size_t)2 * BATCH * NELEM;   // ws + 256 MB
  float* partial = (float*)d_ws;                       // reuse bufA after sort

  potws_sort_kernel<<<2 * BATCH, TPB, 0, stream>>>(x, y, bufA, bufB);
  potws_cost_kernel<<<BATCH, TPB, 0, stream>>>((const float*)bufB, partial);
  potws_final_kernel<<<1, BATCH, 0, stream>>>(partial, (float*)d_out);
}